// LSEPStableLoss_48369921687671
// MI455X (gfx1250) — compile-verified
//
#include <hip/hip_runtime.h>
#include <hip/hip_bf16.h>

typedef __attribute__((ext_vector_type(2))) float v2f;
typedef __attribute__((ext_vector_type(8))) float v8f;

#define LSEP_N 256
#define LSEP_M 512

// One wave (32 lanes) per 16-row tile.
// lane L: row = L & 15, column half = L >> 4 (256 contiguous floats each).
__global__ __launch_bounds__(32)
void lsep_rows_kernel(const float* __restrict__ inp,
                      const int*   __restrict__ tgt,
                      float*       __restrict__ row_lsep) {
  __shared__ float sS0[16];
  __shared__ float sS1[16];

  const int lane = threadIdx.x;      // 0..31
  const int m    = lane & 15;        // row within tile
  const int h    = lane >> 4;        // which half of the columns
  const int r0   = blockIdx.x * 16;  // first row of tile

  const float* sp = inp + (size_t)(r0 + m) * LSEP_M + h * 256;
  const int*   tp = tgt + (size_t)(r0 + m) * LSEP_M + h * 256;
  const float4* sp4 = (const float4*)sp;
  const int4*   tp4 = (const int4*)tp;

  // ---- pass 1: lane-local row max / min over 256 floats ----
  float vmax = -3.402823466e+38f;
  float vmin =  3.402823466e+38f;
#pragma unroll 4
  for (int i = 0; i < 64; ++i) {
    float4 v = sp4[i];
    vmax = fmaxf(vmax, fmaxf(fmaxf(v.x, v.y), fmaxf(v.z, v.w)));
    vmin = fminf(vmin, fminf(fminf(v.x, v.y), fminf(v.z, v.w)));
  }
  // merge the two column halves of each row (partner lane = L ^ 16)
  vmax = fmaxf(vmax, __shfl_xor(vmax, 16, 32));
  vmin = fminf(vmin, __shfl_xor(vmin, 16, 32));
  // maxd = max over all pairwise differences s_j - s_k = rowmax - rowmin
  const float maxd = vmax - vmin;

  // ---- pass 2: masked, stabilized exp partial sums ----
  // S0 = sum_{t==0} exp(s - rowmax),  S1 = sum_{t==1} exp(rowmin - s)
  float e0 = 0.0f, e1 = 0.0f;
#pragma unroll 4
  for (int i = 0; i < 64; ++i) {
    float4 v = sp4[i];
    int4   t = tp4[i];
    e0 += (t.x == 0) ? expf(v.x - vmax) : 0.0f;
    e1 += (t.x != 0) ? expf(vmin - v.x) : 0.0f;
    e0 += (t.y == 0) ? expf(v.y - vmax) : 0.0f;
    e1 += (t.y != 0) ? expf(vmin - v.y) : 0.0f;
    e0 += (t.z == 0) ? expf(v.z - vmax) : 0.0f;
    e1 += (t.z != 0) ? expf(vmin - v.z) : 0.0f;
    e0 += (t.w == 0) ? expf(v.w - vmax) : 0.0f;
    e1 += (t.w != 0) ? expf(vmin - v.w) : 0.0f;
  }

  // ---- WMMA cross-lane reduction: D = A(16x4) x ones(4x16) ----
  // A layout: v0 = K=0 (lanes 0-15) / K=2 (lanes 16-31); v1 = K=1 / K=3.
  // Putting each lane's partial in component 0 (and 0 in component 1) gives
  // D[m,n] = partial_lo[m] + partial_hi[m] = rowsum[m], broadcast over n.
  v2f a0; a0.x = e0; a0.y = 0.0f;
  v2f a1; a1.x = e1; a1.y = 0.0f;
  v2f bones; bones.x = 1.0f; bones.y = 1.0f;
  v8f cz = {};
  v8f d0 = __builtin_amdgcn_wmma_f32_16x16x4_f32(
      false, a0, false, bones, (short)0, cz, false, false);
  v8f d1 = __builtin_amdgcn_wmma_f32_16x16x4_f32(
      false, a1, false, bones, (short)0, cz, false, false);

  // D layout: VGPR i holds M=i (lanes 0-15) and M=8+i (lanes 16-31).
  // Lanes 0 and 16 publish the 16 row sums through LDS.
  if (lane == 0) {
#pragma unroll
    for (int i = 0; i < 8; ++i) { sS0[i] = d0[i]; sS1[i] = d1[i]; }
  }
  if (lane == 16) {
#pragma unroll
    for (int i = 0; i < 8; ++i) { sS0[8 + i] = d0[i]; sS1[8 + i] = d1[i]; }
  }
  __syncthreads();

  if (lane < 16) {
    // lane < 16 holds rowmax/rowmin for row == lane
    float S0 = sS0[lane];
    float S1 = sS1[lane];
    float lsep = maxd + logf(expf(-maxd) + S0 * S1);
    row_lsep[r0 + lane] = lsep;
  }
}

// Deterministic 256 -> 1 mean: fixed per-lane accumulation order + fixed xor tree.
__global__ __launch_bounds__(32)
void lsep_mean_kernel(const float* __restrict__ row_lsep,
                      float*       __restrict__ out) {
  const int lane = threadIdx.x;
  float s = 0.0f;
#pragma unroll
  for (int i = 0; i < LSEP_N / 32; ++i) s += row_lsep[lane + i * 32];
#pragma unroll
  for (int off = 16; off >= 1; off >>= 1) s += __shfl_xor(s, off, 32);
  if (lane == 0) out[0] = s * (1.0f / (float)LSEP_N);
}

extern "C" void kernel_launch(void* const* d_in, const int* in_sizes, int n_in,
                              void* d_out, int out_size, void* d_ws, size_t ws_size,
                              hipStream_t stream) {
  (void)in_sizes; (void)n_in; (void)out_size; (void)ws_size;
  const float* inp = (const float*)d_in[0];
  const int*   tgt = (const int*)d_in[1];
  float* ws  = (float*)d_ws;   // 256 per-row lsep values
  float* out = (float*)d_out;  // single float (mean)

  lsep_rows_kernel<<<LSEP_N / 16, 32, 0, stream>>>(inp, tgt, ws);
  lsep_mean_kernel<<<1, 32, 0, stream>>>(ws, out);
}